// SummaRunner_29506425323848
// MI455X (gfx1250) — compile-verified
//
#include <hip/hip_runtime.h>
#include <math.h>

typedef float v2f __attribute__((ext_vector_type(2)));
typedef float v8f __attribute__((ext_vector_type(8)));
typedef int   v4i_ __attribute__((ext_vector_type(4)));

#define B_    512
#define L_    100
#define D_    1024
#define M_TOT (B_ * L_)   // 51200 rows of the big GEMM

// ---- CDNA5 async global->LDS staging (ASYNCcnt path), with sync fallback ----
#if __has_builtin(__builtin_amdgcn_global_load_async_to_lds_b128)
#define ASYNC_COPY16(lds_ptr, g_ptr)                                         \
  __builtin_amdgcn_global_load_async_to_lds_b128(                            \
      (__attribute__((address_space(1))) v4i_*)(g_ptr),                      \
      (__attribute__((address_space(3))) v4i_*)(lds_ptr), 0, 0)
#if __has_builtin(__builtin_amdgcn_s_wait_asynccnt)
#define ASYNC_WAIT() __builtin_amdgcn_s_wait_asynccnt(0)
#else
#define ASYNC_WAIT() asm volatile("s_wait_asynccnt 0x0" ::: "memory")
#endif
#else
#define ASYNC_COPY16(lds_ptr, g_ptr)                                         \
  do { *(float4*)(lds_ptr) = *(const float4*)(g_ptr); } while (0)
#define ASYNC_WAIT() do { } while (0)
#endif

__device__ __forceinline__ float waveReduce(float v) {
#pragma unroll
  for (int off = 16; off > 0; off >>= 1) v += __shfl_xor(v, off, 32);
  return v;
}

// ---------------------------------------------------------------------------
// Kernel 1: per-batch prep.  docs -> doc_vec -> u -> pre[b,l]
// ---------------------------------------------------------------------------
__global__ __launch_bounds__(256) void prep_kernel(
    const float* __restrict__ sent,      // [B,L,D]
    const float* __restrict__ fc_w,      // [D,D]
    const float* __restrict__ fc_b,      // [D]
    const float* __restrict__ content_w, // [D]
    const float* __restrict__ sal_w,     // [D,D]
    const float* __restrict__ abs_emb,   // [100,50]
    const float* __restrict__ rel_emb,   // [10,50]
    const float* __restrict__ abs_w,     // [50]
    const float* __restrict__ rel_w,     // [50]
    const float* __restrict__ bias,      // [1]
    const int*   __restrict__ doc_lens,  // [B]
    float* __restrict__ pre)             // [B,L]
{
  __shared__ float sDocs[D_];
  __shared__ float sDocV[D_];
  __shared__ float sU[D_];

  const int b    = blockIdx.x;
  const int tid  = threadIdx.x;
  const int len  = doc_lens[b];
  const float lenf = (float)len;
  const float* sb  = sent + (size_t)b * L_ * D_;

  // ---- docs[b] : masked mean over valid sentences (4 dims per thread) ----
  float4 acc = make_float4(0.f, 0.f, 0.f, 0.f);
  for (int l = 0; l < len; ++l) {
    float4 h = *(const float4*)(sb + (size_t)l * D_ + tid * 4);
    acc.x += h.x; acc.y += h.y; acc.z += h.z; acc.w += h.w;
  }
  sDocs[tid * 4 + 0] = acc.x / lenf;
  sDocs[tid * 4 + 1] = acc.y / lenf;
  sDocs[tid * 4 + 2] = acc.z / lenf;
  sDocs[tid * 4 + 3] = acc.w / lenf;
  __syncthreads();

  // ---- doc_vec[e] = tanh(fc_w[e,:] . docs + fc_b[e]) : 4 rows/thread ----
#pragma unroll
  for (int j = 0; j < 4; ++j) {
    const int e = tid * 4 + j;
    const float* wrow = fc_w + (size_t)e * D_;
    float s = 0.f;
    for (int d = 0; d < D_; d += 4) {
      float4 w4 = *(const float4*)(wrow + d);
      s += w4.x * sDocs[d] + w4.y * sDocs[d + 1] +
           w4.z * sDocs[d + 2] + w4.w * sDocs[d + 3];
    }
    sDocV[e] = tanhf(s + fc_b[e]);
  }
  __syncthreads();

  // ---- u[d] = sal_w[d,:] . doc_vec : 4 rows/thread ----
#pragma unroll
  for (int j = 0; j < 4; ++j) {
    const int d = tid * 4 + j;
    const float* wrow = sal_w + (size_t)d * D_;
    float s = 0.f;
    for (int e = 0; e < D_; e += 4) {
      float4 w4 = *(const float4*)(wrow + e);
      s += w4.x * sDocV[e] + w4.y * sDocV[e + 1] +
           w4.z * sDocV[e + 2] + w4.w * sDocV[e + 3];
    }
    sU[d] = s;
  }
  __syncthreads();

  // ---- pre[b,l] = content + sal + abs_p + rel_p + bias (wave per row) ----
  const int wid  = tid >> 5;
  const int lane = tid & 31;
  const float biasv = bias[0];
  for (int l = wid; l < L_; l += 8) {
    const float* hrow = sb + (size_t)l * D_;
    float pc = 0.f, ps = 0.f;
    for (int d = lane * 4; d < D_; d += 128) {
      float4 h  = *(const float4*)(hrow + d);
      float4 cw = *(const float4*)(content_w + d);
      pc += h.x * cw.x + h.y * cw.y + h.z * cw.z + h.w * cw.w;
      ps += h.x * sU[d] + h.y * sU[d + 1] + h.z * sU[d + 2] + h.w * sU[d + 3];
    }
    pc = waveReduce(pc);
    ps = waveReduce(ps);
    if (lane == 0) {
      float ap = 0.f;
      for (int k = 0; k < 50; ++k) ap += abs_emb[l * 50 + k] * abs_w[k];
      int ridx = (int)rintf((float)(l + 1) * 9.0f / lenf);  // RNE == jnp.round
      ridx = ridx < 0 ? 0 : (ridx > 9 ? 9 : ridx);
      float rp = 0.f;
      for (int k = 0; k < 50; ++k) rp += rel_emb[ridx * 50 + k] * rel_w[k];
      pre[b * L_ + l] = pc + ps + ap + rp + biasv;
    }
  }
}

// ---------------------------------------------------------------------------
// Kernel 2: hWn = sent_out(viewed [M_TOT,D]) @ nov_w   via V_WMMA_F32_16X16X4
// Block tile 64(M) x 128(N), K double-buffered 32 at a time via async-to-LDS.
// 8 waves -> 2x4 wave grid, each wave a 32x32 register tile (2x2 of 16x16).
// ---------------------------------------------------------------------------
#define AS_   36    // ldsA row stride (dwords): mult of 4 -> b128-aligned
#define BS_   136   // ldsB row stride (dwords): mult of 4, = 8 mod 32 banks
#define KC_   32    // K chunk

__global__ __launch_bounds__(256) void hwn_gemm_kernel(
    const float* __restrict__ A,   // [M_TOT, D]  sent_out
    const float* __restrict__ Bm,  // [D, D]      nov_w  (k-major rows)
    float* __restrict__ C)         // [M_TOT, D]  hWn
{
  __shared__ float ldsA[2][64 * AS_];   // [m][k]
  __shared__ float ldsB[2][KC_ * BS_];  // [k][n] (untransposed; async b128)

  const int tid  = threadIdx.x;
  const int n0   = blockIdx.x * 128;  // N fastest -> adjacent blocks share A
  const int m0   = blockIdx.y * 64;
  const int wid  = tid >> 5;
  const int lane = tid & 31;
  const int wm   = wid >> 2;          // 0..1  (M 32-stripe)
  const int wn   = wid & 3;           // 0..3  (N 32-stripe)
  const int hi   = lane >> 4;         // K-pair selector (ISA 7.12.2 layout)
  const int lo   = lane & 15;         // row (A) / col (B,C)

  v8f c00 = {}, c01 = {}, c10 = {}, c11 = {};

  // prologue: chunk 0 -> buffer 0  (A: 2 b128/thread, B: 4 b128/thread)
#pragma unroll
  for (int t = 0; t < 2; ++t) {
    const int idx = t * 256 + tid;
    const int mr = idx >> 3, kc4 = (idx & 7) * 4;
    ASYNC_COPY16(&ldsA[0][mr * AS_ + kc4], &A[(size_t)(m0 + mr) * D_ + kc4]);
  }
#pragma unroll
  for (int t = 0; t < 4; ++t) {
    const int idx = t * 256 + tid;
    const int kr = idx >> 5, nc4 = (idx & 31) * 4;
    ASYNC_COPY16(&ldsB[0][kr * BS_ + nc4], &Bm[(size_t)kr * D_ + n0 + nc4]);
  }

  const int NCHUNK = D_ / KC_;
  for (int ch = 0; ch < NCHUNK; ++ch) {
    const int p = ch & 1;
    ASYNC_WAIT();        // this wave's async LDS writes for buffer p done
    __syncthreads();     // all waves' writes done / prev compute done

    if (ch + 1 < NCHUNK) {  // kick off next chunk into the other buffer
      const int kb = (ch + 1) * KC_;
#pragma unroll
      for (int t = 0; t < 2; ++t) {
        const int idx = t * 256 + tid;
        const int mr = idx >> 3, kc4 = (idx & 7) * 4;
        ASYNC_COPY16(&ldsA[p ^ 1][mr * AS_ + kc4],
                     &A[(size_t)(m0 + mr) * D_ + kb + kc4]);
      }
#pragma unroll
      for (int t = 0; t < 4; ++t) {
        const int idx = t * 256 + tid;
        const int kr = idx >> 5, nc4 = (idx & 31) * 4;
        ASYNC_COPY16(&ldsB[p ^ 1][kr * BS_ + nc4],
                     &Bm[(size_t)(kb + kr) * D_ + n0 + nc4]);
      }
    }

    const float* la = ldsA[p];
    const float* lb = ldsB[p];
#pragma unroll
    for (int ks = 0; ks < KC_ / 4; ++ks) {
      const int kk = ks * 4 + 2 * hi;
      // A frags (b64, conflict-free): lane lo = row, {kk,kk+1} K pair
      v2f a0 = *(const v2f*)&la[(wm * 32 + lo) * AS_ + kk];
      v2f a1 = *(const v2f*)&la[(wm * 32 + 16 + lo) * AS_ + kk];
      // B frags (2x b32 each): lane lo = col, rows kk / kk+1
      v2f b0, b1;
      b0.x = lb[kk * BS_ + wn * 32 + lo];
      b0.y = lb[(kk + 1) * BS_ + wn * 32 + lo];
      b1.x = lb[kk * BS_ + wn * 32 + 16 + lo];
      b1.y = lb[(kk + 1) * BS_ + wn * 32 + 16 + lo];
      c00 = __builtin_amdgcn_wmma_f32_16x16x4_f32(false, a0, false, b0,
                                                  (short)0, c00, false, false);
      c01 = __builtin_amdgcn_wmma_f32_16x16x4_f32(false, a0, false, b1,
                                                  (short)0, c01, false, false);
      c10 = __builtin_amdgcn_wmma_f32_16x16x4_f32(false, a1, false, b0,
                                                  (short)0, c10, false, false);
      c11 = __builtin_amdgcn_wmma_f32_16x16x4_f32(false, a1, false, b1,
                                                  (short)0, c11, false, false);
    }
  }

  // store D: vgpr v -> rows v (lanes 0-15) / 8+v (lanes 16-31), col = lo
  const int colBase = n0 + wn * 32 + lo;
#pragma unroll
  for (int v = 0; v < 8; ++v) {
    const int r0 = m0 + wm * 32 + hi * 8 + v;
    const int r1 = r0 + 16;
    C[(size_t)r0 * D_ + colBase]      = c00[v];
    C[(size_t)r0 * D_ + colBase + 16] = c01[v];
    C[(size_t)r1 * D_ + colBase]      = c10[v];
    C[(size_t)r1 * D_ + colBase + 16] = c11[v];
  }
}

// ---------------------------------------------------------------------------
// Kernel 3: sequential novelty scan per batch.  s lives in registers (f4/thr).
// ---------------------------------------------------------------------------
__global__ __launch_bounds__(256) void scan_kernel(
    const float* __restrict__ sent,     // [B,L,D]
    const float* __restrict__ hWn,      // [B,L,D]
    const float* __restrict__ pre,      // [B,L]
    const int*   __restrict__ doc_lens, // [B]
    float* __restrict__ out)            // [B,L]
{
  __shared__ float red[9];
  const int b    = blockIdx.x;
  const int tid  = threadIdx.x;
  const int wid  = tid >> 5;
  const int lane = tid & 31;
  const int len  = doc_lens[b];
  const size_t base = (size_t)b * L_ * D_;

  float4 s = make_float4(0.f, 0.f, 0.f, 0.f);

  for (int l = 0; l < L_; ++l) {
    const size_t off = base + (size_t)l * D_ + tid * 4;
    float4 h = *(const float4*)(sent + off);
    float4 w = *(const float4*)(hWn + off);
    if (l + 1 < L_) {  // overlap next row fetch with the reduction
      __builtin_prefetch(hWn + off + D_, 0, 0);
      __builtin_prefetch(sent + off + D_, 0, 0);
    }

    float4 t;
    t.x = tanhf(s.x); t.y = tanhf(s.y); t.z = tanhf(s.z); t.w = tanhf(s.w);
    float part = w.x * t.x + w.y * t.y + w.z * t.z + w.w * t.w;
    part = waveReduce(part);
    if (lane == 0) red[wid] = part;
    __syncthreads();
    if (tid == 0) {
      float tot = 0.f;
#pragma unroll
      for (int i = 0; i < 8; ++i) tot += red[i];
      red[8] = tot;
    }
    __syncthreads();
    const float novelty = -red[8];
    const float prob =
        (l < len) ? 1.0f / (1.0f + expf(-(pre[b * L_ + l] + novelty))) : 0.0f;
    if (tid == 0) out[b * L_ + l] = prob;

    s.x += prob * h.x; s.y += prob * h.y;
    s.z += prob * h.z; s.w += prob * h.w;
    __syncthreads();  // protect red[] before next iteration's writes
  }
}

// ---------------------------------------------------------------------------
extern "C" void kernel_launch(void* const* d_in, const int* in_sizes, int n_in,
                              void* d_out, int out_size, void* d_ws, size_t ws_size,
                              hipStream_t stream) {
  const float* sent      = (const float*)d_in[0];
  const float* fc_w      = (const float*)d_in[1];
  const float* fc_b      = (const float*)d_in[2];
  const float* content_w = (const float*)d_in[3];
  const float* sal_w     = (const float*)d_in[4];
  const float* nov_w     = (const float*)d_in[5];
  const float* abs_emb   = (const float*)d_in[6];
  const float* rel_emb   = (const float*)d_in[7];
  const float* abs_w     = (const float*)d_in[8];
  const float* rel_w     = (const float*)d_in[9];
  const float* bias      = (const float*)d_in[10];
  const int*   doc_lens  = (const int*)d_in[11];
  float* out = (float*)d_out;

  float* hWn = (float*)d_ws;                                  // 200 MB
  float* pre = (float*)((char*)d_ws +
                        (size_t)M_TOT * D_ * sizeof(float));  // 200 KB

  prep_kernel<<<B_, 256, 0, stream>>>(sent, fc_w, fc_b, content_w, sal_w,
                                      abs_emb, rel_emb, abs_w, rel_w, bias,
                                      doc_lens, pre);
  hwn_gemm_kernel<<<dim3(D_ / 128, M_TOT / 64), 256, 0, stream>>>(sent, nov_w,
                                                                  hWn);
  scan_kernel<<<B_, 256, 0, stream>>>(sent, hWn, pre, doc_lens, out);
}